// LabelSmoothingLossMoE_27367531610372
// MI455X (gfx1250) — compile-verified
//
#include <hip/hip_runtime.h>
#include <hip/hip_bf16.h>
#include <stdint.h>

// ---------------- problem constants (from reference) ----------------
#define VSIZE    32000
#define NEXP     8
#define SMOOTH   0.1f
#define CONF     0.9f

// ---------------- tiling ----------------
#define BLK      256                      // 8 waves of 32
#define TILE_F4  1024                     // 4096 floats = 16 KB per LDS buffer
#define TILE_B   (TILE_F4 * 16)           // 16384 bytes
#define TOTAL_F4 (VSIZE / 4)              // 8000 float4 per row
#define NTILES   ((TOTAL_F4 + TILE_F4 - 1) / TILE_F4)   // 8 (7 full + 1 ragged)
#define ITERS    (TILE_F4 / BLK)          // 4 async issues per wave per tile

__device__ __forceinline__ void softmerge(float& m, float& s, float m2, float s2) {
    float nm = fmaxf(m, m2);
    s = s * __expf(m - nm) + s2 * __expf(m2 - nm);
    m = nm;
}

// Full tile: SADDR form. INST_OFFSET is added to BOTH the LDS and global
// addresses (CDNA5 ISA), so the 4 issues need zero per-issue
// address math: fixed vLDS = ldsbase + tid*16, fixed vOFF = tid*16,
// saddr = rowbase + tile*16384 (uniform SGPR pair).
#define ISSUE_FULL(_tile, _nb)                                                      \
    do {                                                                            \
        unsigned long long _sa = rowbase + (unsigned long long)(_tile) * TILE_B;    \
        asm volatile(                                                               \
            "global_load_async_to_lds_b128 %0, %1, %2 offset:0 th:TH_LOAD_NT\n\t"   \
            "global_load_async_to_lds_b128 %0, %1, %2 offset:4096 th:TH_LOAD_NT\n\t"\
            "global_load_async_to_lds_b128 %0, %1, %2 offset:8192 th:TH_LOAD_NT\n\t"\
            "global_load_async_to_lds_b128 %0, %1, %2 offset:12288 th:TH_LOAD_NT"   \
            :: "v"(la[(_nb)]), "v"(vb), "s"(_sa) : "memory");                       \
    } while (0)

// Ragged last tile: per-lane clamped 64-bit-address form so every wave still
// issues exactly ITERS instructions (keeps ASYNCcnt accounting uniform).
#define ISSUE_LAST(_nb)                                                             \
    do {                                                                            \
        unsigned _lds0 = ldsbase[(_nb)];                                            \
        _Pragma("unroll")                                                           \
        for (int _it = 0; _it < ITERS; ++_it) {                                     \
            int _slot = tid + _it * BLK;                                            \
            int _f4   = (NTILES - 1) * TILE_F4 + _slot;                             \
            _f4 = (_f4 < TOTAL_F4) ? _f4 : 0;                                       \
            unsigned long long _ga = rowbase + (unsigned long long)(_f4) * 16;      \
            unsigned _la = _lds0 + (unsigned)(_slot * 16);                          \
            asm volatile("global_load_async_to_lds_b128 %0, %1, off th:TH_LOAD_NT"  \
                         :: "v"(_la), "v"(_ga) : "memory");                         \
        }                                                                           \
    } while (0)

#define WAIT_ASYNC_LE4  asm volatile("s_wait_asynccnt 4" ::: "memory")
#define WAIT_ASYNC_LE0  asm volatile("s_wait_asynccnt 0" ::: "memory")

// -------- kernel 1: per-row streaming online-softmax reduction --------
__global__ __launch_bounds__(BLK)
void row_kl_kernel(const float* __restrict__ x,
                   const int*   __restrict__ target,
                   float*       __restrict__ rowkl) {
    const int row = blockIdx.x;
    const int tid = threadIdx.x;

    const int tgt = target[row];
    if (tgt == 0) {                       // PAD_IDX row: contributes nothing
        if (tid == 0) rowkl[row] = 0.0f;
        return;
    }

    __shared__ float4 buf[2][TILE_F4];    // 2 x 16 KB double buffer
    __shared__ float  wm[BLK / 32], wsum[BLK / 32], wx[BLK / 32];

    const unsigned long long rowbase =
        (unsigned long long)(uintptr_t)(x + (size_t)row * VSIZE);
    const unsigned vb = (unsigned)(tid * 16);
    unsigned ldsbase[2] = { (unsigned)(uintptr_t)&buf[0][0],
                            (unsigned)(uintptr_t)&buf[1][0] };
    unsigned la[2] = { ldsbase[0] + vb, ldsbase[1] + vb };

    float xt = 0.0f;
    if (tid == 0) xt = x[(size_t)row * VSIZE + tgt];   // early, hides latency

    ISSUE_FULL(0, 0);

    float m = -3.0e38f, s = 0.0f, sx = 0.0f;

    for (int t = 0; t < NTILES; ++t) {
        const int nb = t & 1;
        if (t + 1 < NTILES - 1)      { ISSUE_FULL(t + 1, (t + 1) & 1); WAIT_ASYNC_LE4; }
        else if (t + 1 == NTILES - 1){ ISSUE_LAST((t + 1) & 1);        WAIT_ASYNC_LE4; }
        else                         { WAIT_ASYNC_LE0; }

        if (t < NTILES - 1) {
            // full tile: one (m,s) merge per 16 elements -> ~1.06 exp/elem
            float4 v0 = buf[nb][tid];
            float4 v1 = buf[nb][tid + 256];
            float4 v2 = buf[nb][tid + 512];
            float4 v3 = buf[nb][tid + 768];
            float m0 = fmaxf(fmaxf(v0.x, v0.y), fmaxf(v0.z, v0.w));
            float m1 = fmaxf(fmaxf(v1.x, v1.y), fmaxf(v1.z, v1.w));
            float m2 = fmaxf(fmaxf(v2.x, v2.y), fmaxf(v2.z, v2.w));
            float m3 = fmaxf(fmaxf(v3.x, v3.y), fmaxf(v3.z, v3.w));
            float vm = fmaxf(fmaxf(m0, m1), fmaxf(m2, m3));
            float nm = fmaxf(m, vm);
            float acc =
                ((__expf(v0.x - nm) + __expf(v0.y - nm)) + (__expf(v0.z - nm) + __expf(v0.w - nm))) +
                ((__expf(v1.x - nm) + __expf(v1.y - nm)) + (__expf(v1.z - nm) + __expf(v1.w - nm))) +
                ((__expf(v2.x - nm) + __expf(v2.y - nm)) + (__expf(v2.z - nm) + __expf(v2.w - nm))) +
                ((__expf(v3.x - nm) + __expf(v3.y - nm)) + (__expf(v3.z - nm) + __expf(v3.w - nm)));
            s = s * __expf(m - nm) + acc;
            m = nm;
            sx += (((v0.x + v0.y) + (v0.z + v0.w)) + ((v1.x + v1.y) + (v1.z + v1.w))) +
                  (((v2.x + v2.y) + (v2.z + v2.w)) + ((v3.x + v3.y) + (v3.z + v3.w)));
        } else {
            // ragged last tile: per-float4 guarded online merge
#pragma unroll
            for (int it = 0; it < ITERS; ++it) {
                int slot = tid + it * BLK;
                int f4   = t * TILE_F4 + slot;
                if (f4 < TOTAL_F4) {
                    float4 v  = buf[nb][slot];
                    float  vm = fmaxf(fmaxf(v.x, v.y), fmaxf(v.z, v.w));
                    float  nm = fmaxf(m, vm);
                    s = s * __expf(m - nm) +
                        ((__expf(v.x - nm) + __expf(v.y - nm)) +
                         (__expf(v.z - nm) + __expf(v.w - nm)));
                    m = nm;
                    sx += (v.x + v.y) + (v.z + v.w);
                }
            }
        }
    }

    // intra-wave reduction (wave32)
#pragma unroll
    for (int off = 16; off > 0; off >>= 1) {
        float m2 = __shfl_down(m,  off, 32);
        float s2 = __shfl_down(s,  off, 32);
        float x2 = __shfl_down(sx, off, 32);
        softmerge(m, s, m2, s2);
        sx += x2;
    }
    const int wave = tid >> 5, lane = tid & 31;
    if (lane == 0) { wm[wave] = m; wsum[wave] = s; wx[wave] = sx; }
    __syncthreads();

    if (tid == 0) {
        float M = wm[0], S = wsum[0], X = wx[0];
#pragma unroll
        for (int w = 1; w < BLK / 32; ++w) { softmerge(M, S, wm[w], wsum[w]); X += wx[w]; }

        const float lse      = M + __logf(S);
        const float sum_logp = X - (float)VSIZE * lse;
        const float logp_t   = xt - lse;

        const float sv = SMOOTH / (float)(VSIZE - 1);
        const float C0 = (float)(VSIZE - 1) * sv * __logf(sv) + CONF * __logf(CONF);

        rowkl[row] = C0 - sv * sum_logp - (CONF - sv) * logp_t;
    }
}

// -------- kernel 2: tiny reductions + final combine (single block) --------
__global__ __launch_bounds__(BLK)
void finalize_kernel(const float* __restrict__ topkv,
                     const float* __restrict__ gate,
                     const int*   __restrict__ target,
                     const int*   __restrict__ topki,
                     const float* __restrict__ rowkl,
                     float*       __restrict__ out,
                     int nrows, int mtop) {
    const int tid = threadIdx.x;
    __shared__ float rKL[BLK], rZ[BLK], rC[BLK];
    __shared__ float hl[BLK][NEXP], hs[BLK][NEXP];

    float kl = 0.0f, z = 0.0f, cnt = 0.0f;
    for (int r = tid; r < nrows; r += BLK) {
        kl  += rowkl[r];
        cnt += (target[r] != 0) ? 1.0f : 0.0f;
        const float* g = gate + (size_t)r * NEXP;
        float mg = g[0];
#pragma unroll
        for (int e = 1; e < NEXP; ++e) mg = fmaxf(mg, g[e]);
        float sg = 0.0f;
#pragma unroll
        for (int e = 0; e < NEXP; ++e) sg += __expf(g[e] - mg);
        float lse = mg + __logf(sg);
        z += lse * lse;
    }
    rKL[tid] = kl; rZ[tid] = z; rC[tid] = cnt;

    float el[NEXP], es[NEXP];
#pragma unroll
    for (int e = 0; e < NEXP; ++e) { el[e] = 0.0f; es[e] = 0.0f; }
    for (int i = tid; i < mtop; i += BLK) {
        int e = topki[i] & (NEXP - 1);
        el[e] += 1.0f;
        es[e] += topkv[i];
    }
#pragma unroll
    for (int e = 0; e < NEXP; ++e) { hl[tid][e] = el[e]; hs[tid][e] = es[e]; }
    __syncthreads();

    for (int str = BLK / 2; str > 0; str >>= 1) {   // fixed-order tree: deterministic
        if (tid < str) {
            rKL[tid] += rKL[tid + str];
            rZ[tid]  += rZ[tid + str];
            rC[tid]  += rC[tid + str];
#pragma unroll
            for (int e = 0; e < NEXP; ++e) {
                hl[tid][e] += hl[tid + str][e];
                hs[tid][e] += hs[tid + str][e];
            }
        }
        __syncthreads();
    }

    if (tid == 0) {
        float load = 0.0f;
#pragma unroll
        for (int e = 0; e < NEXP; ++e) load += hl[0][e] * hs[0][e];
        load *= (float)NEXP / (float)nrows;

        float denom = (rC[0] > 0.0f) ? rC[0] : 1.0f;
        out[0] = rKL[0] / denom + 0.01f * load + 0.001f * (rZ[0] / (float)nrows);
    }
}

extern "C" void kernel_launch(void* const* d_in, const int* in_sizes, int n_in,
                              void* d_out, int out_size, void* d_ws, size_t ws_size,
                              hipStream_t stream) {
    const float* x      = (const float*)d_in[0];   // [N, 32000]
    const float* topkv  = (const float*)d_in[1];   // [N, K]
    const float* gate   = (const float*)d_in[2];   // [N, 8]
    const int*   target = (const int*)  d_in[3];   // [N]
    const int*   topki  = (const int*)  d_in[4];   // [N, K]
    float*       out    = (float*)d_out;
    float*       rowkl  = (float*)d_ws;            // N floats of scratch

    const int nrows = in_sizes[3];                 // 4096
    const int mtop  = in_sizes[4];                 // 8192

    row_kl_kernel<<<nrows, BLK, 0, stream>>>(x, target, rowkl);
    finalize_kernel<<<1, BLK, 0, stream>>>(topkv, gate, target, topki, rowkl, out,
                                           nrows, mtop);
}